// FLAttention_19645180412781
// MI455X (gfx1250) — compile-verified
//
#include <hip/hip_runtime.h>
#include <math.h>

typedef __attribute__((ext_vector_type(2))) float v2f;
typedef __attribute__((ext_vector_type(8))) float v8f;

#define B_DIM 64
#define D_DIM 512
#define H_DIM 8
#define EPS_F 1e-8f

// Only probe for the device builtin in the device compilation pass; the host
// pass has no amdgcn builtins and would trip the #error spuriously.
#if defined(__HIP_DEVICE_COMPILE__) && defined(__AMDGCN__)
#if !__has_builtin(__builtin_amdgcn_wmma_f32_16x16x4_f32)
#error "missing __builtin_amdgcn_wmma_f32_16x16x4_f32 on this toolchain (device pass)"
#endif
#endif

__global__ void fl_init_out(const float* __restrict__ x, float* __restrict__ out, int n) {
    int i = blockIdx.x * blockDim.x + threadIdx.x;
    if (i < n) out[i] = x[i];
}

__global__ __launch_bounds__(256) void fl_attn_kernel(const float* __restrict__ x,
                                                      const float* __restrict__ alphas,
                                                      const float* __restrict__ betas,
                                                      float* __restrict__ out) {
    __shared__ float sQ[D_DIM];
    __shared__ float sK[D_DIM];
    __shared__ float sV[D_DIM];
    __shared__ float sC[8][16 * 16];   // per-wave C spill for layout remap

    const int hb  = blockIdx.x;
    const int b   = hb & (B_DIM - 1);
    const int h   = hb >> 6;           // hb / B_DIM
    const int tid = threadIdx.x;

    const float aq = alphas[h * 3 + 0], ak = alphas[h * 3 + 1], av = alphas[h * 3 + 2];
    const float bq = betas[h * 3 + 0],  bk = betas[h * 3 + 1],  bv = betas[h * 3 + 2];

    // q, k, v rows for this (h, b) into LDS
    for (int j = tid; j < D_DIM; j += 256) {
        float xv = x[b * D_DIM + j];
        sQ[j] = fmaf(aq, xv, bq);
        sK[j] = fmaf(ak, xv, bk);
        sV[j] = fmaf(av, xv, bv);
    }
    __syncthreads();

    const int wave  = tid >> 5;
    const int lane  = tid & 31;
    const int half  = lane >> 4;       // 0: lanes 0-15, 1: lanes 16-31
    const int laneM = lane & 15;       // A-matrix row within tile; B/C column N
    const int kb    = half * 2;        // K base within the 4-wide chunk
    const float invSqrtD = 0.044194173824159216f;   // 1/sqrt(512)

    // 32 row-tiles of 16; wave w handles tiles w, w+8, w+16, w+24 (uniform per wave)
    for (int t = wave; t < 32; t += 8) {
        const int i0 = t * 16;
        const float ki = sK[i0 + laneM];

        // ---- pass 1: min_j |q_j - k_i|  (halves scan disjoint j ranges, combine) ----
        float mn = 3.402823466e+38f;
        const int jbeg = half * 256;
        #pragma unroll 8
        for (int j = jbeg; j < jbeg + 256; ++j) {
            mn = fminf(mn, fabsf(sQ[j] - ki));
        }
        mn = fminf(mn, __shfl_xor(mn, 16));
        const float smax = __builtin_amdgcn_rcpf(mn + EPS_F);   // max_j s_ij

        // ---- pass 2: e_ij = exp(s_ij - smax); WMMA-reduce Σe and Σe*v jointly ----
        // A (16x4 f32): lane<16 holds {K=0,1}, lane>=16 holds {K=2,3}, row = laneM.
        // B (4x16 f32): col 0 = 1.0 (denominator), col 1 = v[j+K] (numerator), else 0.
        v8f c0 = {};
        v8f c1 = {};
        for (int j = 0; j < D_DIM; j += 8) {
            {
                const int jj = j + kb;
                float q0 = sQ[jj], q1 = sQ[jj + 1];
                float w0 = sV[jj], w1 = sV[jj + 1];
                float s0 = __builtin_amdgcn_rcpf(fabsf(q0 - ki) + EPS_F);
                float s1 = __builtin_amdgcn_rcpf(fabsf(q1 - ki) + EPS_F);
                float e0 = __expf(s0 - smax);
                float e1 = __expf(s1 - smax);
                float b0 = (laneM == 0) ? 1.0f : ((laneM == 1) ? w0 : 0.0f);
                float b1 = (laneM == 0) ? 1.0f : ((laneM == 1) ? w1 : 0.0f);
                v2f A  = {e0, e1};
                v2f Bm = {b0, b1};
                c0 = __builtin_amdgcn_wmma_f32_16x16x4_f32(false, A, false, Bm,
                                                           (short)0, c0, false, false);
            }
            {
                const int jj = j + 4 + kb;
                float q0 = sQ[jj], q1 = sQ[jj + 1];
                float w0 = sV[jj], w1 = sV[jj + 1];
                float s0 = __builtin_amdgcn_rcpf(fabsf(q0 - ki) + EPS_F);
                float s1 = __builtin_amdgcn_rcpf(fabsf(q1 - ki) + EPS_F);
                float e0 = __expf(s0 - smax);
                float e1 = __expf(s1 - smax);
                float b0 = (laneM == 0) ? 1.0f : ((laneM == 1) ? w0 : 0.0f);
                float b1 = (laneM == 0) ? 1.0f : ((laneM == 1) ? w1 : 0.0f);
                v2f A  = {e0, e1};
                v2f Bm = {b0, b1};
                c1 = __builtin_amdgcn_wmma_f32_16x16x4_f32(false, A, false, Bm,
                                                           (short)0, c1, false, false);
            }
        }

        // ---- extract: C layout is (VGPR r, lane N) -> remap via per-wave LDS ----
        const int M0 = half * 8;
        #pragma unroll
        for (int r = 0; r < 8; ++r) {
            sC[wave][(M0 + r) * 16 + laneM] = c0[r] + c1[r];
        }
        asm volatile("s_wait_dscnt 0" ::: "memory");   // wave-local store->load ordering
        if (lane < 16) {
            float sumE  = sC[wave][lane * 16 + 0];
            float sumEV = sC[wave][lane * 16 + 1];
            float att   = (sumEV / sumE) * invSqrtD;
            atomicAdd(&out[b * D_DIM + i0 + lane], att);
        }
    }
}

extern "C" void kernel_launch(void* const* d_in, const int* in_sizes, int n_in,
                              void* d_out, int out_size, void* d_ws, size_t ws_size,
                              hipStream_t stream) {
    const float* x      = (const float*)d_in[0];
    const float* alphas = (const float*)d_in[1];
    const float* betas  = (const float*)d_in[2];
    float* out          = (float*)d_out;

    const int n = B_DIM * D_DIM;                   // 32768
    fl_init_out<<<(n + 255) / 256, 256, 0, stream>>>(x, out, n);
    fl_attn_kernel<<<H_DIM * B_DIM, 256, 0, stream>>>(x, alphas, betas, out);
}